// CriticLoss_23141283791036
// MI455X (gfx1250) — compile-verified
//
#include <hip/hip_runtime.h>
#include <math.h>

// CriticLoss for MI455X (gfx1250, wave32).
// Bandwidth-bound: 64MiB logits streamed once (~2.9us floor @ 23.3TB/s).
// predicted = probs . bin_centers computed on the matrix pipe with
// V_WMMA_F32_16X16X4_F32 (full fp32 WMMA), 16 rows per wave per group.

#define NUM_BINS 128
#define VMIN_F  (-150.0f)
#define VMAX_F  (150.0f)
#define BW_F    (300.0f / 127.0f)
#define INV_BW  (127.0f / 300.0f)

#define WAVES_PER_BLOCK 4
#define ROWPAD 132   // 128 + 4 pad -> LDS rows land on distinct banks (132 % 64 == 4)

typedef __attribute__((ext_vector_type(2))) float v2f;
typedef __attribute__((ext_vector_type(8))) float v8f;

__device__ __forceinline__ float wave_max(float v) {
#pragma unroll
  for (int off = 16; off >= 1; off >>= 1)
    v = fmaxf(v, __shfl_xor(v, off, 32));
  return v;
}
__device__ __forceinline__ float wave_sum(float v) {
#pragma unroll
  for (int off = 16; off >= 1; off >>= 1)
    v += __shfl_xor(v, off, 32);
  return v;
}

__global__ __launch_bounds__(WAVES_PER_BLOCK * 32)
void critic_main(const float* __restrict__ logits,
                 const float* __restrict__ target,
                 const float* __restrict__ mask,
                 float* __restrict__ part,   // [5][gridDim.x]
                 int n_groups, int n_rows) {
  __shared__ float lds_probs[WAVES_PER_BLOCK][16][ROWPAD];
  __shared__ float red[WAVES_PER_BLOCK][5];

  const int lane  = threadIdx.x & 31;
  const int wave  = threadIdx.x >> 5;
  const int gwave = blockIdx.x * WAVES_PER_BLOCK + wave;
  const int nwave = gridDim.x * WAVES_PER_BLOCK;

  float ce_acc = 0.f, mse_acc = 0.f, pred_acc = 0.f, tgt_acc = 0.f, msk_acc = 0.f;

  float (*P)[ROWPAD] = lds_probs[wave];

  for (int g = gwave; g < n_groups; g += nwave) {   // wave-uniform loop
    const int row0 = g * 16;

    // ---- Phase 1: per-row softmax stats + CE (wave32 shuffle reductions) ----
    for (int r = 0; r < 16; ++r) {
      const int  row   = row0 + r;
      const bool valid = (row < n_rows);
      const int  crow  = valid ? row : (n_rows - 1);

      // 4 bins per lane, coalesced 512B per row across the wave
      const float4 x = *(const float4*)(logits + (size_t)crow * NUM_BINS + lane * 4);

      float m = fmaxf(fmaxf(x.x, x.y), fmaxf(x.z, x.w));
      m = wave_max(m);

      const float e0 = __expf(x.x - m), e1 = __expf(x.y - m);
      const float e2 = __expf(x.z - m), e3 = __expf(x.w - m);
      const float denom = wave_sum(e0 + e1 + e2 + e3);
      const float inv   = 1.0f / denom;
      const float L     = __logf(denom);   // log_prob[k] = x[k] - m - L

      // stage probs into LDS in WMMA-B staging tile
      float4 p; p.x = e0 * inv; p.y = e1 * inv; p.z = e2 * inv; p.w = e3 * inv;
      *(float4*)&P[r][lane * 4] = p;

      // two-hot CE: each lane contributes only for bins it owns
      const float tv  = target[crow];
      const float mk  = valid ? mask[crow] : 0.f;
      const float v   = fminf(fmaxf(tv, VMIN_F), VMAX_F);
      const float nrm = (v - VMIN_F) * INV_BW;
      int lo = (int)floorf(nrm); lo = min(max(lo, 0), NUM_BINS - 1);
      int hi = (int)ceilf(nrm);  hi = min(max(hi, 0), NUM_BINS - 1);
      const float uw = nrm - (float)lo;
      const float lw = 1.0f - uw;

      const int base = lane * 4;
      const int lb = lo - base, hb = hi - base;
      float contrib = 0.f;
      if (lb >= 0 && lb < 4) {
        const float xl = (lb == 0) ? x.x : (lb == 1) ? x.y : (lb == 2) ? x.z : x.w;
        contrib += lw * (xl - m - L);
      }
      if (hb >= 0 && hb < 4) {
        const float xh = (hb == 0) ? x.x : (hb == 1) ? x.y : (hb == 2) ? x.z : x.w;
        contrib += uw * (xh - m - L);
      }
      ce_acc -= contrib * mk;   // ce = -(lw*lp_lo + uw*lp_up), mask-weighted
    }

    // DS ops complete in-order within a wave on CDNA5; just stop the compiler
    // from reordering the LDS reads above the writes.
    asm volatile("" ::: "memory");

    // ---- Phase 2: predicted[n] = sum_k probs[n][k]*centers[k] via fp32 WMMA ----
    // D = A x B + C, 16x16x4 fp32. A[m,k] = centers (replicated over m),
    // B[k,n] = probs of data-row n. After 32 K-chunks, D[0,n] = predicted[n].
    // A layout (ISA): lanes 0-15 hold K=0,1 ; lanes 16-31 hold K=2,3.
    // B layout (mirrored): lanes 0-15: N=lane, K=0,1 ; lanes 16-31: N=lane-16, K=2,3.
    v8f acc = {0.f, 0.f, 0.f, 0.f, 0.f, 0.f, 0.f, 0.f};
    const int ksel = (lane < 16) ? 0 : 2;
    const int n    = lane & 15;
#pragma unroll
    for (int kc = 0; kc < 32; ++kc) {
      const int kb = kc * 4 + ksel;
      v2f a, b;
      a.x = VMIN_F + (float)(kb)     * BW_F;
      a.y = VMIN_F + (float)(kb + 1) * BW_F;
      const float2 bp = *(const float2*)&P[n][kb];   // conflict-free (stride 132)
      b.x = bp.x; b.y = bp.y;
      acc = __builtin_amdgcn_wmma_f32_16x16x4_f32(
          /*neg_a=*/false, a, /*neg_b=*/false, b,
          /*c_mod=*/(short)0, acc, /*reuse_a=*/false, /*reuse_b=*/false);
    }

    // D[0, N=lane] sits in accumulator VGPR0 of lanes 0..15
    {
      const float pred  = acc[0];
      const int   row   = row0 + n;
      const bool  valid = (row < n_rows);
      const int   crow  = valid ? row : (n_rows - 1);
      const float tv = target[crow];
      const float mk = valid ? mask[crow] : 0.f;
      const float pm = pred * mk, tm = tv * mk, d = pm - tm;
      if (lane < 16) {
        mse_acc  += d * d;
        pred_acc += pm;
        tgt_acc  += tm;
        msk_acc  += mk;
      }
    }
    asm volatile("" ::: "memory");  // keep LDS reuse ordered across groups
  }

  // ---- deterministic block reduction ----
  ce_acc   = wave_sum(ce_acc);
  mse_acc  = wave_sum(mse_acc);
  pred_acc = wave_sum(pred_acc);
  tgt_acc  = wave_sum(tgt_acc);
  msk_acc  = wave_sum(msk_acc);
  if (lane == 0) {
    red[wave][0] = ce_acc;  red[wave][1] = mse_acc; red[wave][2] = pred_acc;
    red[wave][3] = tgt_acc; red[wave][4] = msk_acc;
  }
  __syncthreads();
  if (threadIdx.x == 0) {
    float s[5] = {0.f, 0.f, 0.f, 0.f, 0.f};
    for (int w = 0; w < WAVES_PER_BLOCK; ++w)
      for (int c = 0; c < 5; ++c) s[c] += red[w][c];
    for (int c = 0; c < 5; ++c) part[c * gridDim.x + blockIdx.x] = s[c];
  }
}

__global__ __launch_bounds__(256)
void critic_finalize(const float* __restrict__ part, float* __restrict__ out, int nb) {
  __shared__ float sm[5][256];
  const int t = threadIdx.x;
  float a[5] = {0.f, 0.f, 0.f, 0.f, 0.f};
  for (int i = t; i < nb; i += 256)
    for (int c = 0; c < 5; ++c) a[c] += part[c * nb + i];
  for (int c = 0; c < 5; ++c) sm[c][t] = a[c];
  __syncthreads();
  for (int s = 128; s >= 1; s >>= 1) {
    if (t < s)
      for (int c = 0; c < 5; ++c) sm[c][t] += sm[c][t + s];
    __syncthreads();
  }
  if (t == 0) {
    const float ms = fmaxf(sm[4][0], 1.0f);
    out[0] = sm[0][0] / ms;   // critic_loss
    out[1] = sm[1][0] / ms;   // critic_mse
    out[2] = sm[2][0] / ms;   // predicted_mean
    out[3] = sm[3][0] / ms;   // target_mean
  }
}

extern "C" void kernel_launch(void* const* d_in, const int* in_sizes, int n_in,
                              void* d_out, int out_size, void* d_ws, size_t ws_size,
                              hipStream_t stream) {
  const float* logits = (const float*)d_in[0];   // [B,S,128] f32
  const float* target = (const float*)d_in[1];   // [B,S] f32
  const float* mask   = (const float*)d_in[2];   // [B,S] f32
  float* out = (float*)d_out;                    // 4 scalars

  const int n_rows   = in_sizes[1];
  const int n_groups = (n_rows + 15) / 16;

  int nb = 256;
  if ((size_t)nb * 5 * sizeof(float) > ws_size) {
    nb = (int)(ws_size / (5 * sizeof(float)));
    if (nb < 1) nb = 1;
  }
  float* part = (float*)d_ws;

  critic_main<<<nb, WAVES_PER_BLOCK * 32, 0, stream>>>(logits, target, mask,
                                                       part, n_groups, n_rows);
  critic_finalize<<<1, 256, 0, stream>>>(part, out, nb);
}